// MultiStageResidualVQ_67791763800756
// MI455X (gfx1250) — compile-verified
//
#include <hip/hip_runtime.h>
#include <math.h>

typedef float v2f __attribute__((ext_vector_type(2)));
typedef float v8f __attribute__((ext_vector_type(8)));

#define N_ROWS   32768      // B*Q = 8*4096
#define DDIM     256
#define KCODES   1024
#define DEPTH    3
#define NELEM    8388608    // N_ROWS * DDIM
#define ROWS_PER_BLOCK 128
#define LDS_STRIDE 260      // 256 + 4 pad: 16B-aligned rows, conflict-free frag reads
#define CHUNK    32         // codes per inner chunk (two 16-wide tiles per wave)
#define NCHUNKS  (KCODES / CHUNK)

// d_out layout (floats): q_total[8388608], vq_loss[1], composed[32768], perps[3]
#define OUT_LOSS 8388608
#define OUT_COMP 8388609
#define OUT_PERP (8388609 + 32768)

// ---------------- async-to-LDS helpers (CDNA5) ----------------
#if defined(__has_builtin)
#if __has_builtin(__builtin_amdgcn_global_load_async_to_lds_b128)
#define USE_ASYNC_LDS 1
#endif
#endif

// b128 transfer unit: builtin prototype wants int-vector pointees
typedef int vi4 __attribute__((__vector_size__(4 * sizeof(int))));
typedef __attribute__((address_space(1))) vi4 GVi4;   // global (__device__) AS
typedef __attribute__((address_space(3))) vi4 LVi4;   // LDS (__shared__) AS

__device__ __forceinline__ void copy16_to_lds(const float* g, float* l) {
#if defined(USE_ASYNC_LDS)
    // flat-shared low 32 bits are the LDS byte address (aperture rule), so the
    // integer round-trip matches what addrspacecast generic->AS3 lowers to.
    __builtin_amdgcn_global_load_async_to_lds_b128(
        (GVi4*)(uintptr_t)g, (LVi4*)(unsigned)(uintptr_t)l, 0, 0);
#else
    *(float4*)l = *(const float4*)g;
#endif
}

__device__ __forceinline__ void async_wait0() {
#if defined(USE_ASYNC_LDS)
    asm volatile("s_wait_asynccnt 0x0" ::: "memory");
#endif
}

// ---------------- init kernels ----------------

__global__ void k_zero_counts(unsigned* __restrict__ counts) {
    int t = threadIdx.x;
    for (int i = t; i < DEPTH * KCODES; i += 256) counts[i] = 0u;
}

__global__ void k_e2(const float* __restrict__ cbs, float* __restrict__ e2s) {
    int t = blockIdx.x * 256 + threadIdx.x;           // 0..3071 (12 blocks)
    if (t >= DEPTH * KCODES) return;
    const float4* row = (const float4*)(cbs + (size_t)t * DDIM);
    float s = 0.f;
    for (int i = 0; i < DDIM / 4; ++i) {
        float4 v = row[i];
        s += v.x * v.x + v.y * v.y + v.z * v.z + v.w * v.w;
    }
    e2s[t] = s;
}

__global__ void k_copy(const float* __restrict__ src, float* __restrict__ dst) {
    int e = blockIdx.x * 256 + threadIdx.x;           // float4 index, 8192 blocks
    ((float4*)dst)[e] = ((const float4*)src)[e];
}

// ---------------- WMMA argmin kernel ----------------
// block: 256 threads = 8 waves; 128 rows per block; 32 chunks of 32 codes.
// Each wave owns 16 rows and accumulates a 16x32 f32 score tile (2 C-tiles)
// via 2x64 v_wmma_f32_16x16x4_f32 per chunk; one A-frag feeds both tiles.
// B staging is double-buffered with async global->LDS copies overlapping WMMA.

__global__ void k_argmin(const float* __restrict__ residual,
                         const float* __restrict__ cb,
                         const float* __restrict__ e2s,
                         int* __restrict__ idx_out) {
    extern __shared__ float lds[];
    float* Alds = lds;                                    // 128 x LDS_STRIDE
    float* B0   = lds + ROWS_PER_BLOCK * LDS_STRIDE;      // 32 x LDS_STRIDE
    float* B1   = B0 + CHUNK * LDS_STRIDE;                // 32 x LDS_STRIDE

    const int tid  = threadIdx.x;
    const int lane = tid & 31;
    const int wave = tid >> 5;                 // 0..7
    const int m    = lane & 15;                // row/col within 16
    const int grp  = lane >> 4;                // 0/1
    const int koff = grp * 2;                  // A/B fragment K-pair select
    const int rowBase = blockIdx.x * ROWS_PER_BLOCK;

    // stage A tile (128 x 256 f32) into LDS
    for (int f = tid; f < ROWS_PER_BLOCK * (DDIM / 4); f += 256) {
        int row = f >> 6;                      // 64 float4 per row
        int c4  = f & 63;
        copy16_to_lds(residual + (size_t)(rowBase + row) * DDIM + (c4 << 2),
                      &Alds[row * LDS_STRIDE + (c4 << 2)]);
    }
    // stage B chunk 0
    for (int f = tid; f < CHUNK * (DDIM / 4); f += 256) {
        int crow = f >> 6;
        int c4   = f & 63;
        copy16_to_lds(cb + (size_t)crow * DDIM + (c4 << 2),
                      &B0[crow * LDS_STRIDE + (c4 << 2)]);
    }
    async_wait0();
    __syncthreads();

    float bestS[8];
    int   bestC[8];
#pragma unroll
    for (int i = 0; i < 8; ++i) { bestS[i] = 3.4e38f; bestC[i] = 0; }

    const float* Abase = &Alds[(wave * 16 + m) * LDS_STRIDE + koff];

    for (int chunk = 0; chunk < NCHUNKS; ++chunk) {
        float* cur = (chunk & 1) ? B1 : B0;
        float* nxt = (chunk & 1) ? B0 : B1;

        // kick off async staging of the next chunk into the other buffer
        // (that buffer was last read in the previous iteration; the barrier at the
        //  end of the previous iteration guarantees everyone is done with it)
        if (chunk + 1 < NCHUNKS) {
            const float* src = cb + (size_t)(chunk + 1) * CHUNK * DDIM;
            for (int f = tid; f < CHUNK * (DDIM / 4); f += 256) {
                int crow = f >> 6;
                int c4   = f & 63;
                copy16_to_lds(src + (size_t)crow * DDIM + (c4 << 2),
                              &nxt[crow * LDS_STRIDE + (c4 << 2)]);
            }
        }

        v8f acc0 = {0.f, 0.f, 0.f, 0.f, 0.f, 0.f, 0.f, 0.f};
        v8f acc1 = {0.f, 0.f, 0.f, 0.f, 0.f, 0.f, 0.f, 0.f};
        const float* Bb0 = &cur[m * LDS_STRIDE + koff];
        const float* Bb1 = &cur[(16 + m) * LDS_STRIDE + koff];
#pragma unroll
        for (int kk = 0; kk < DDIM / 4; ++kk) {
            v2f a  = *(const v2f*)(Abase + kk * 4);
            v2f b0 = *(const v2f*)(Bb0 + kk * 4);
            v2f b1 = *(const v2f*)(Bb1 + kk * 4);
            acc0 = __builtin_amdgcn_wmma_f32_16x16x4_f32(
                false, a, false, b0, (short)0, acc0, false, false);
            acc1 = __builtin_amdgcn_wmma_f32_16x16x4_f32(
                false, a, false, b1, (short)0, acc1, false, false);
        }

        const int   code0 = (chunk << 5) + m;     // tile-0 column/code id
        const float e0 = 0.5f * e2s[code0];
        const float e1 = 0.5f * e2s[code0 + 16];

#pragma unroll
        for (int i = 0; i < 8; ++i) {
            // acc[i] = S[row = i + 8*grp][code]; dist ~ 0.5*e2 - S
            float s = e0 - acc0[i];
            int   c = code0;
#pragma unroll
            for (int off = 1; off < 16; off <<= 1) {
                float os = __shfl_xor(s, off, 32);
                int   oc = __shfl_xor(c, off, 32);
                if (os < s || (os == s && oc < c)) { s = os; c = oc; }
            }
            if (s < bestS[i]) { bestS[i] = s; bestC[i] = c; }

            s = e1 - acc1[i];
            c = code0 + 16;
#pragma unroll
            for (int off = 1; off < 16; off <<= 1) {
                float os = __shfl_xor(s, off, 32);
                int   oc = __shfl_xor(c, off, 32);
                if (os < s || (os == s && oc < c)) { s = os; c = oc; }
            }
            if (s < bestS[i]) { bestS[i] = s; bestC[i] = c; }
        }

        async_wait0();      // next buffer fully landed in LDS (this wave's part)
        __syncthreads();    // everyone's part landed; everyone done reading cur
    }

    if (m == 0) {
        int rbase = rowBase + wave * 16 + grp * 8;
#pragma unroll
        for (int i = 0; i < 8; ++i) idx_out[rbase + i] = bestC[i];
    }
}

// ---------------- per-stage bookkeeping ----------------

__global__ void k_hist(const int* __restrict__ idx, unsigned* __restrict__ counts) {
    int i = blockIdx.x * 256 + threadIdx.x;    // 128 blocks
    if (i < N_ROWS) atomicAdd(&counts[idx[i]], 1u);
}

// residual -= cb[idx]; per-block sum of squares -> lossPart[block] (no float atomics)
__global__ void k_update(const float* __restrict__ cb,
                         const int* __restrict__ idx,
                         float* __restrict__ residual,
                         float* __restrict__ lossPart) {
    __shared__ float red[256];
    int e4  = blockIdx.x * 256 + threadIdx.x;  // float4 index, 8192 blocks
    int row = e4 >> 6;
    int c4  = e4 & 63;
    int id  = idx[row];
    float4 r = ((const float4*)residual)[e4];
    float4 e = ((const float4*)cb)[id * (DDIM / 4) + c4];
    r.x -= e.x; r.y -= e.y; r.z -= e.z; r.w -= e.w;
    ((float4*)residual)[e4] = r;
    red[threadIdx.x] = r.x * r.x + r.y * r.y + r.z * r.z + r.w * r.w;
    __syncthreads();
    for (int s = 128; s > 0; s >>= 1) {
        if (threadIdx.x < s) red[threadIdx.x] += red[threadIdx.x + s];
        __syncthreads();
    }
    if (threadIdx.x == 0) lossPart[blockIdx.x] = red[0];
}

// ---------------- finalize ----------------

__global__ void k_qtotal(const float* __restrict__ z,
                         const float* __restrict__ residual,
                         float* __restrict__ out) {
    int e4 = blockIdx.x * 256 + threadIdx.x;   // 8192 blocks
    float4 a = ((const float4*)z)[e4];
    float4 b = ((const float4*)residual)[e4];
    float4 q; q.x = a.x - b.x; q.y = a.y - b.y; q.z = a.z - b.z; q.w = a.w - b.w;
    ((float4*)out)[e4] = q;
}

__global__ void k_compose(const int* __restrict__ indices, float* __restrict__ out) {
    int i = blockIdx.x * 256 + threadIdx.x;    // 128 blocks
    if (i < N_ROWS) {
        long long v = (long long)indices[i]
                    + ((long long)indices[N_ROWS + i]     << 10)
                    + ((long long)indices[2 * N_ROWS + i] << 20);
        out[i] = (float)v;
    }
}

__global__ void k_final(const unsigned* __restrict__ counts,
                        const float* __restrict__ lossPart,
                        float* __restrict__ out) {
    __shared__ float red[256];
    int tid = threadIdx.x;

    // vq_loss = 1.25 * (sum of all squared residual deltas) / NELEM
    float acc = 0.f;
    for (int i = tid; i < DEPTH * 8192; i += 256) acc += lossPart[i];
    red[tid] = acc; __syncthreads();
    for (int s = 128; s > 0; s >>= 1) {
        if (tid < s) red[tid] += red[tid + s];
        __syncthreads();
    }
    if (tid == 0) out[OUT_LOSS] = red[0] * 1.25f / (float)NELEM;
    __syncthreads();

    for (int st = 0; st < DEPTH; ++st) {
        float h = 0.f;
        for (int c = tid; c < KCODES; c += 256) {
            float p = (float)counts[st * KCODES + c] * (1.0f / (float)N_ROWS);
            h += p * logf(p + 1e-10f);
        }
        red[tid] = h; __syncthreads();
        for (int s = 128; s > 0; s >>= 1) {
            if (tid < s) red[tid] += red[tid + s];
            __syncthreads();
        }
        if (tid == 0) out[OUT_PERP + st] = expf(-red[0]);
        __syncthreads();
    }
}

// ---------------- launcher ----------------

extern "C" void kernel_launch(void* const* d_in, const int* in_sizes, int n_in,
                              void* d_out, int out_size, void* d_ws, size_t ws_size,
                              hipStream_t stream) {
    const float* z   = (const float*)d_in[0];   // (8,4096,256) f32
    const float* cbs = (const float*)d_in[1];   // (3,1024,256) f32
    float* out = (float*)d_out;

    // workspace carve-up
    char* ws = (char*)d_ws;
    float*    residual = (float*)ws;                         ws += (size_t)NELEM * 4;
    int*      indices  = (int*)ws;                           ws += (size_t)DEPTH * N_ROWS * 4;
    unsigned* counts   = (unsigned*)ws;                      ws += (size_t)DEPTH * KCODES * 4;
    float*    e2s      = (float*)ws;                         ws += (size_t)DEPTH * KCODES * 4;
    float*    lossPart = (float*)ws;                         // DEPTH * 8192 floats

    const size_t ldsBytes =
        (size_t)(ROWS_PER_BLOCK + 2 * CHUNK) * LDS_STRIDE * sizeof(float); // ~195 KB

    k_zero_counts<<<1, 256, 0, stream>>>(counts);
    k_e2<<<12, 256, 0, stream>>>(cbs, e2s);
    k_copy<<<NELEM / 4 / 256, 256, 0, stream>>>(z, residual);

    for (int d = 0; d < DEPTH; ++d) {
        const float* cb = cbs + (size_t)d * KCODES * DDIM;
        int* idx = indices + (size_t)d * N_ROWS;
        k_argmin<<<N_ROWS / ROWS_PER_BLOCK, 256, ldsBytes, stream>>>(
            residual, cb, e2s + d * KCODES, idx);
        k_hist<<<N_ROWS / 256, 256, 0, stream>>>(idx, counts + d * KCODES);
        k_update<<<NELEM / 4 / 256, 256, 0, stream>>>(
            cb, idx, residual, lossPart + (size_t)d * 8192);
    }

    k_qtotal<<<NELEM / 4 / 256, 256, 0, stream>>>(z, residual, out);
    k_compose<<<N_ROWS / 256, 256, 0, stream>>>(indices, out + OUT_COMP);
    k_final<<<1, 256, 0, stream>>>(counts, lossPart, out);
}